// EdgeNetwork_28630251995174
// MI455X (gfx1250) — compile-verified
//
#include <hip/hip_runtime.h>
#include <stdint.h>

// EdgeNetwork: out[b,e] = sigmoid( tanh( concat(X[b,Ro[b,e]], X[b,Ri[b,e]]) @ W1 + b1 ) @ W2 + b2 )
// B=8, N=50000, E=200000, D=64, H=8.
// Gather-bound (~850 MB random reads, X fits in 192MB L2). MLP done with
// V_WMMA_F32_16X16X4_F32: one wave32 per 16-edge tile, K=128 in 32 chunks of 4.

typedef float v2f __attribute__((ext_vector_type(2)));
typedef float v8f __attribute__((ext_vector_type(8)));

#define D_      64
#define H_      8
#define KCHUNKS 32   // 128 / 4

__device__ __forceinline__ float sigmoid_f(float x) {
    return 1.0f / (1.0f + __expf(-x));
}

__global__ __launch_bounds__(256) void edge_mlp_wmma(
    const float* __restrict__ X,        // [B,N,64]
    const long long* __restrict__ Ri,   // [B,E] int64
    const long long* __restrict__ Ro,   // [B,E] int64
    const float* __restrict__ W1,       // [128,8] row-major
    const float* __restrict__ b1,       // [8]
    const float* __restrict__ W2,       // [8]
    const float* __restrict__ b2,       // [1]
    float* __restrict__ out,            // [B,E]
    int Bn, int Nn, int En)
{
    const int lane = threadIdx.x & 31;
    const int col  = lane & 15;   // output row (edge-in-tile) for A, N-col for B/C
    const int half = lane >> 4;   // K sub-slot within a K=4 chunk

    // ---- Preload W1 as 32 B-tiles (4x16, cols 8..15 zero) into VGPRs; reused
    //      across the whole persistent grid-stride loop.
    // B layout (4x16 f32, mirrors A 16x4): lanes 0-15 hold K=0,1 for N=lane,
    // lanes 16-31 hold K=2,3 for N=lane-16.
    v2f Bm[KCHUNKS];
    #pragma unroll
    for (int kc = 0; kc < KCHUNKS; ++kc) {
        const int krow = kc * 4 + half * 2;
        float w0 = 0.0f, w1v = 0.0f;
        if (col < H_) {
            w0  = W1[(long long)krow * H_ + col];
            w1v = W1[(long long)(krow + 1) * H_ + col];
        }
        v2f bm = {w0, w1v};
        Bm[kc] = bm;
    }
    const float b1v = (col < H_) ? b1[col] : 0.0f;
    const float w2v = (col < H_) ? W2[col] : 0.0f;
    const float b2v = b2[0];

    const int tilesPerBatch = (En + 15) >> 4;
    const long long nTiles  = (long long)Bn * tilesPerBatch;
    const long long wave    = (long long)blockIdx.x * (blockDim.x >> 5) + (threadIdx.x >> 5);
    const long long nWaves  = (long long)gridDim.x * (blockDim.x >> 5);

    // Wave-uniform loop bound -> EXEC stays all-1s through every WMMA.
    for (long long tile = wave; tile < nTiles; tile += nWaves) {
        const int b  = (int)(tile / tilesPerBatch);
        const int e0 = (int)(tile % tilesPerBatch) << 4;

        // Each lane owns edge-row `col` (both lane-halves mirror the same rows).
        int e = e0 + col;
        if (e >= En) e = En - 1;                 // clamp partial tile (store masked later)
        const long long ei = (long long)b * En + e;
        const long long ro = Ro[ei];
        const long long ri = Ri[ei];
        const float* po = X + ((long long)b * Nn + ro) * D_;   // bo: features 0..63
        const float* pi = X + ((long long)b * Nn + ri) * D_;   // bi: features 64..127

        // ---- C = A(16x128) x W1(128x16pad), K in chunks of 4 via f32 WMMA.
        v8f c = {};
        #pragma unroll
        for (int kc = 0; kc < KCHUNKS; ++kc) {
            // A layout: lane holds K = kc*4 + half*2 + {0,1} of its row.
            const float* src = (kc < 16) ? (po + kc * 4 + half * 2)
                                         : (pi + (kc - 16) * 4 + half * 2);
            v2f a = *(const v2f*)src;            // global_load_b64, 8B-aligned
            c = __builtin_amdgcn_wmma_f32_16x16x4_f32(
                    /*neg_a=*/false, a, /*neg_b=*/false, Bm[kc],
                    /*c_mod=*/(short)0, c, /*reuse_a=*/false, /*reuse_b=*/false);
        }

        // ---- Layer 2: per C-VGPR j, value is h[row = j + half*8, col].
        // t = tanh(c + b1[col]) * W2[col]; butterfly-reduce the 8 hidden cols
        // inside each 8-lane group (cols 8..15 contribute exactly 0).
        float red[8];
        #pragma unroll
        for (int j = 0; j < 8; ++j) {
            float t = tanhf(c[j] + b1v) * w2v;
            t += __shfl_xor(t, 1, 32);
            t += __shfl_xor(t, 2, 32);
            t += __shfl_xor(t, 4, 32);
            red[j] = t;       // lanes 0..7: row j ; lanes 16..23: row j+8
        }

        // Transpose 16 row-sums into lanes 0..15, one value per lane.
        float val = 0.0f;
        #pragma unroll
        for (int r = 0; r < 8; ++r) {
            const float sA = __shfl(red[r], 0, 32);   // row r
            const float sB = __shfl(red[r], 16, 32);  // row r + 8
            val = (lane == r)     ? sA : val;
            val = (lane == r + 8) ? sB : val;
        }

        const int eo = e0 + lane;
        if (lane < 16 && eo < En) {
            out[(long long)b * En + eo] = sigmoid_f(val + b2v);
        }
    }
}

extern "C" void kernel_launch(void* const* d_in, const int* in_sizes, int n_in,
                              void* d_out, int out_size, void* d_ws, size_t ws_size,
                              hipStream_t stream) {
    const float*     X  = (const float*)d_in[0];
    const long long* Ri = (const long long*)d_in[1];
    const long long* Ro = (const long long*)d_in[2];
    const float*     W1 = (const float*)d_in[3];
    const float*     b1 = (const float*)d_in[4];
    const float*     W2 = (const float*)d_in[5];
    const float*     b2 = (const float*)d_in[6];
    float*           out = (float*)d_out;

    const int Bn = 8;
    const int En = in_sizes[1] / Bn;               // 200000
    const int Nn = in_sizes[0] / (Bn * D_);        // 50000

    // Persistent waves: 2048 blocks x 8 waves = 16384 waves over 100000 tiles,
    // amortizing the W1 VGPR preload ~6x.
    dim3 grid(2048), block(256);
    edge_mlp_wmma<<<grid, block, 0, stream>>>(X, Ri, Ro, W1, b1, W2, b2, out, Bn, Nn, En);
}